// GCN_66357244723265
// MI455X (gfx1250) — compile-verified
//
#include <hip/hip_runtime.h>
#include <hip/hip_bf16.h>
#include <math.h>

typedef __attribute__((ext_vector_type(2))) float v2f;
typedef __attribute__((ext_vector_type(8))) float v8f;

#define CDIV(a, b) (((a) + (b) - 1) / (b))

__device__ __forceinline__ void atomAddF32(float* p, float v) {
    (void)__hip_atomic_fetch_add(p, v, __ATOMIC_RELAXED, __HIP_MEMORY_SCOPE_AGENT);
}

// ---------------- degree / normalization ----------------

__global__ void k_deg_init(unsigned* __restrict__ deg, long n) {
    long i = (long)blockIdx.x * blockDim.x + threadIdx.x;
    if (i < n) deg[i] = 1u;  // self-loop contributes 1
}

__global__ void k_deg_count(const int* __restrict__ col, unsigned* __restrict__ deg, long e) {
    long i = (long)blockIdx.x * blockDim.x + threadIdx.x;
    if (i < e) {
        int c = __builtin_nontemporal_load(col + i);
        atomicAdd(&deg[c], 1u);  // global_atomic_add_u32
    }
}

__global__ void k_dis(const unsigned* __restrict__ deg, float* __restrict__ dis, long n) {
    long i = (long)blockIdx.x * blockDim.x + threadIdx.x;
    if (i < n) {
        unsigned d = deg[i];  // same buffer, per-element read-then-write
        dis[i] = 1.0f / sqrtf((float)d);
    }
}

// ---------------- layer 1 linear: h1[N,8] = (x[N,3] @ W1[3,8]) * dis[node]  via WMMA f32 16x16x4
// One wave per 16-node tile. A: 16x4 (K col 3 = 0). B: 4x16 (W1 padded, cols 8..15 = 0).
// A layout: lanes 0-15 -> M=lane, VGPR0/1 = K0,K1 ; lanes 16-31 -> M=lane-16, VGPR0/1 = K2,K3.

__global__ void k_linear1(const float* __restrict__ x, const float* __restrict__ W1,
                          const float* __restrict__ dis, float* __restrict__ h1) {
    int wave = (int)((blockIdx.x * blockDim.x + threadIdx.x) >> 5);  // tile id
    int lane = threadIdx.x & 31;
    int m  = lane & 15;
    int kh = (lane >> 4) * 2;  // 0 or 2
    long node = (long)wave * 16 + m;
    const float* xp = x + node * 3;

    v2f a;
    a.x = (kh < 3) ? xp[kh] : 0.0f;
    a.y = (kh + 1 < 3) ? xp[kh + 1] : 0.0f;

    v2f b;
    b.x = (kh < 3 && m < 8) ? W1[kh * 8 + m] : 0.0f;
    b.y = (kh + 1 < 3 && m < 8) ? W1[(kh + 1) * 8 + m] : 0.0f;

    v8f c = {};
    c = __builtin_amdgcn_wmma_f32_16x16x4_f32(false, a, false, b, (short)0, c, false, false);

    // D: vgpr v, lanes 0-15 -> D[v][lane]; lanes 16-31 -> D[v+8][lane-16]
    if (m < 8) {
        long base = (long)wave * 16 + ((lane >= 16) ? 8 : 0);
#pragma unroll
        for (int v = 0; v < 8; ++v) h1[(base + v) * 8 + m] = c[v] * dis[base + v];
    }
}

// ---------------- layer 2 linear: g[N,2] = (h[N,8] @ W2[8,2]) * dis[node]  via two chained WMMAs

__global__ void k_linear2(const float* __restrict__ h, const float* __restrict__ W2,
                          const float* __restrict__ dis, float* __restrict__ g) {
    int wave = (int)((blockIdx.x * blockDim.x + threadIdx.x) >> 5);
    int lane = threadIdx.x & 31;
    int m  = lane & 15;
    int kh = (lane >> 4) * 2;
    long node = (long)wave * 16 + m;
    const float* hp = h + node * 8;

    v2f a0; a0.x = hp[kh];     a0.y = hp[kh + 1];      // K = 0..3
    v2f a1; a1.x = hp[4 + kh]; a1.y = hp[5 + kh];      // K = 4..7

    v2f b0, b1;
    b0.x = (m < 2) ? W2[(0 + kh) * 2 + m] : 0.0f;
    b0.y = (m < 2) ? W2[(1 + kh) * 2 + m] : 0.0f;
    b1.x = (m < 2) ? W2[(4 + kh) * 2 + m] : 0.0f;
    b1.y = (m < 2) ? W2[(5 + kh) * 2 + m] : 0.0f;

    v8f c = {};
    c = __builtin_amdgcn_wmma_f32_16x16x4_f32(false, a0, false, b0, (short)0, c, false, false);
    c = __builtin_amdgcn_wmma_f32_16x16x4_f32(false, a1, false, b1, (short)0, c, false, false);

    if (m < 2) {
        long base = (long)wave * 16 + ((lane >= 16) ? 8 : 0);
#pragma unroll
        for (int v = 0; v < 8; ++v) g[(base + v) * 2 + m] = c[v] * dis[base + v];
    }
}

// ---------------- aggregation (8 features): out1[c] = h'[c] + sum_{r->c} h'[r] ----------------

__global__ void k_agg8_init(const float* __restrict__ h1, float* __restrict__ out1, long n) {
    long i = (long)blockIdx.x * blockDim.x + threadIdx.x;
    if (i >= n) return;
    const float4* hp = (const float4*)(h1 + i * 8);
    float4* op = (float4*)(out1 + i * 8);
    op[0] = hp[0];
    op[1] = hp[1];
}

__global__ void k_agg8_edges(const int* __restrict__ row, const int* __restrict__ col,
                             const float* __restrict__ h1, float* __restrict__ out1, long e) {
    long i = (long)blockIdx.x * blockDim.x + threadIdx.x;
    if (i >= e) return;
    int r = __builtin_nontemporal_load(row + i);
    int c = __builtin_nontemporal_load(col + i);
    const float4* hp = (const float4*)(h1 + (long)r * 8);
    float4 a = hp[0], b = hp[1];
    float* op = out1 + (long)c * 8;
    atomAddF32(op + 0, a.x);
    atomAddF32(op + 1, a.y);
    atomAddF32(op + 2, a.z);
    atomAddF32(op + 3, a.w);
    atomAddF32(op + 4, b.x);
    atomAddF32(op + 5, b.y);
    atomAddF32(op + 6, b.z);
    atomAddF32(op + 7, b.w);
}

// h[i,f] = relu(dis[i]*out1[i,f] + b1[f])
__global__ void k_bias_relu(const float* __restrict__ out1, const float* __restrict__ b1,
                            const float* __restrict__ dis, float* __restrict__ h, long n) {
    long i = (long)blockIdx.x * blockDim.x + threadIdx.x;
    if (i >= n) return;
    float d = dis[i];
    const float4* ip = (const float4*)(out1 + i * 8);
    float4 a = ip[0], b = ip[1];
    float4* op = (float4*)(h + i * 8);
    op[0] = make_float4(fmaxf(fmaf(a.x, d, b1[0]), 0.0f), fmaxf(fmaf(a.y, d, b1[1]), 0.0f),
                        fmaxf(fmaf(a.z, d, b1[2]), 0.0f), fmaxf(fmaf(a.w, d, b1[3]), 0.0f));
    op[1] = make_float4(fmaxf(fmaf(b.x, d, b1[4]), 0.0f), fmaxf(fmaf(b.y, d, b1[5]), 0.0f),
                        fmaxf(fmaf(b.z, d, b1[6]), 0.0f), fmaxf(fmaf(b.w, d, b1[7]), 0.0f));
}

// ---------------- aggregation (2 features) ----------------

__global__ void k_agg2_init(const float* __restrict__ g, float* __restrict__ out2, long n) {
    long i = (long)blockIdx.x * blockDim.x + threadIdx.x;
    if (i >= n) return;
    ((float2*)(out2 + i * 2))[0] = ((const float2*)(g + i * 2))[0];
}

__global__ void k_agg2_edges(const int* __restrict__ row, const int* __restrict__ col,
                             const float* __restrict__ g, float* __restrict__ out2, long e) {
    long i = (long)blockIdx.x * blockDim.x + threadIdx.x;
    if (i >= e) return;
    int r = __builtin_nontemporal_load(row + i);
    int c = __builtin_nontemporal_load(col + i);
    float2 v = ((const float2*)(g + (long)r * 2))[0];
    float* op = out2 + (long)c * 2;
    atomAddF32(op + 0, v.x);
    atomAddF32(op + 1, v.y);
}

__global__ void k_logsoftmax(const float* __restrict__ out2, const float* __restrict__ b2,
                             const float* __restrict__ dis, float* __restrict__ out, long n) {
    long i = (long)blockIdx.x * blockDim.x + threadIdx.x;
    if (i >= n) return;
    float d = dis[i];
    float2 v = ((const float2*)(out2 + i * 2))[0];
    float a = fmaf(v.x, d, b2[0]);
    float b = fmaf(v.y, d, b2[1]);
    float mx = fmaxf(a, b);
    float mn = fminf(a, b);
    float lse = mx + log1pf(expf(mn - mx));
    out[i * 2 + 0] = a - lse;
    out[i * 2 + 1] = b - lse;
}

// ---------------- launcher ----------------

extern "C" void kernel_launch(void* const* d_in, const int* in_sizes, int n_in,
                              void* d_out, int out_size, void* d_ws, size_t ws_size,
                              hipStream_t stream) {
    const float* x  = (const float*)d_in[0];
    const int*   ei = (const int*)d_in[1];
    const float* W1 = (const float*)d_in[2];
    const float* b1 = (const float*)d_in[3];
    const float* W2 = (const float*)d_in[4];
    const float* b2 = (const float*)d_in[5];
    float* out = (float*)d_out;

    const long N = (long)in_sizes[0] / 3;
    const long E = (long)in_sizes[1] / 2;
    const int* row = ei;       // edge_index[0] = source
    const int* col = ei + E;   // edge_index[1] = target

    // workspace layout (floats): dis[N] | h1[8N] | out1[8N] | g[2N] | out2[2N]
    float* ws   = (float*)d_ws;
    float* dis  = ws;
    float* h1   = ws + N;
    float* out1 = ws + 9 * N;
    float* g    = ws + 17 * N;
    float* out2 = ws + 19 * N;

    const int B = 256;
    const unsigned gN = (unsigned)CDIV(N, B);
    const unsigned gE = (unsigned)CDIV(E, B);
    // WMMA kernels: 1 tile (16 nodes) per wave, 4 waves (128 thr) per block
    const unsigned gT = (unsigned)CDIV(N / 16, 4);

    // degree -> dis (deg counted as u32 in the dis buffer, then converted in place)
    unsigned* deg = (unsigned*)dis;
    k_deg_init<<<gN, B, 0, stream>>>(deg, N);
    k_deg_count<<<gE, B, 0, stream>>>(col, deg, E);
    k_dis<<<gN, B, 0, stream>>>(deg, dis, N);

    // layer 1: h1 = (x@W1)*dis ; out1 = h1 + scatter(h1) ; h1 = relu(dis*out1 + b1)
    k_linear1<<<gT, 128, 0, stream>>>(x, W1, dis, h1);
    k_agg8_init<<<gN, B, 0, stream>>>(h1, out1, N);
    k_agg8_edges<<<gE, B, 0, stream>>>(row, col, h1, out1, E);
    k_bias_relu<<<gN, B, 0, stream>>>(out1, b1, dis, h1, N);

    // layer 2: g = (h1@W2)*dis ; out2 = g + scatter(g) ; out = logsoftmax(dis*out2 + b2)
    k_linear2<<<gT, 128, 0, stream>>>(h1, W2, dis, g);
    k_agg2_init<<<gN, B, 0, stream>>>(g, out2, N);
    k_agg2_edges<<<gE, B, 0, stream>>>(row, col, g, out2, E);
    k_logsoftmax<<<gN, B, 0, stream>>>(out2, b2, dis, out, N);
}